// GATClassifier_18708877542132
// MI455X (gfx1250) — compile-verified
//
#include <hip/hip_runtime.h>
#include <hip/hip_bf16.h>

typedef float v2f __attribute__((ext_vector_type(2)));
typedef float v8f __attribute__((ext_vector_type(8)));

#define NEG_SLOPE 0.2f

// ---------------------------------------------------------------------------
// WMMA GEMM (fp32): C[M,Nout] = A[M,K] @ B[K,Nout]
// One wave32 computes a 16x64 output tile with V_WMMA_F32_16X16X4_F32.
// Requires M%16==0, K%4==0, Nout%64==0 (true for this workload).
// A 16x4 fp32 fragment layout (ISA 7.12.2): lanes 0-15 hold K=0,1 (v0,v1),
// lanes 16-31 hold K=2,3, row = lane&15. B 4x16 symmetric (col = lane&15).
// ---------------------------------------------------------------------------
__global__ __launch_bounds__(128) void gemm_wmma_f32(
    const float* __restrict__ A, const float* __restrict__ B,
    float* __restrict__ C, int M, int K, int Nout) {
  const int lane  = threadIdx.x & 31;
  const int wave  = threadIdx.x >> 5;
  const int tileM = (blockIdx.x * (blockDim.x >> 5) + wave) * 16;
  if (tileM >= M) return;                       // wave-uniform: EXEC stays full
  const int tileN = blockIdx.y * 64;
  const int r  = lane & 15;                     // A row in tile / B,C col in tile
  const int kh = (lane >> 4) * 2;               // K sub-offset for this half-wave

  v8f acc0 = {}, acc1 = {}, acc2 = {}, acc3 = {};
  const float* Arow = A + (size_t)(tileM + r) * K;
  for (int k = 0; k < K; k += 4) {
    v2f a;
    a.x = Arow[k + kh];
    a.y = Arow[k + kh + 1];
    const float* B0 = B + (size_t)(k + kh) * Nout + tileN + r;
    const float* B1 = B0 + Nout;
    v2f b0, b1, b2, b3;
    b0.x = B0[0];  b0.y = B1[0];
    b1.x = B0[16]; b1.y = B1[16];
    b2.x = B0[32]; b2.y = B1[32];
    b3.x = B0[48]; b3.y = B1[48];
    acc0 = __builtin_amdgcn_wmma_f32_16x16x4_f32(false, a, false, b0, (short)0, acc0, false, false);
    acc1 = __builtin_amdgcn_wmma_f32_16x16x4_f32(false, a, false, b1, (short)0, acc1, false, false);
    acc2 = __builtin_amdgcn_wmma_f32_16x16x4_f32(false, a, false, b2, (short)0, acc2, false, false);
    acc3 = __builtin_amdgcn_wmma_f32_16x16x4_f32(false, a, false, b3, (short)0, acc3, false, false);
  }
  const int crow = tileM + (lane >> 4) * 8;     // C: VGPR i -> row i (+8 for hi half)
  const int ccol = tileN + (lane & 15);
  for (int i = 0; i < 8; ++i) {
    float* Cp = C + (size_t)(crow + i) * Nout + ccol;
    Cp[0]  = acc0[i];
    Cp[16] = acc1[i];
    Cp[32] = acc2[i];
    Cp[48] = acc3[i];
  }
}

// ------------------------- monotone float<->uint ---------------------------
__device__ __forceinline__ unsigned ord_f32(float f) {
  unsigned u = __float_as_uint(f);
  return (u & 0x80000000u) ? ~u : (u | 0x80000000u);
}
__device__ __forceinline__ float unord_f32(unsigned u) {
  return (u & 0x80000000u) ? __uint_as_float(u & 0x7FFFFFFFu) : __uint_as_float(~u);
}

__device__ __forceinline__ void edge_endpoints(const int* __restrict__ ei,
                                               int E, int e, int& s, int& d) {
  if (e < E) { s = ei[e]; d = ei[E + e]; }
  else       { s = e - E; d = e - E; }          // appended self-loops
}

// s_src[n,h] = <h[n,h,:], a_src[h,:]> ; likewise s_dst
__global__ __launch_bounds__(256) void scores_kernel(
    const float* __restrict__ h, const float* __restrict__ a_src,
    const float* __restrict__ a_dst, float* __restrict__ ssrc,
    float* __restrict__ sdst, int N, int H, int C) {
  int idx = blockIdx.x * blockDim.x + threadIdx.x;
  if (idx >= N * H) return;
  int hh = idx % H;
  const float* hp = h + (size_t)idx * C;
  const float* as = a_src + hh * C;
  const float* ad = a_dst + hh * C;
  float s0 = 0.f, s1 = 0.f;
  for (int c = 0; c < C; ++c) { float v = hp[c]; s0 += v * as[c]; s1 += v * ad[c]; }
  ssrc[idx] = s0;
  sdst[idx] = s1;
}

// Pass 1: e = leaky_relu(s_src[src]+s_dst[dst]); store; segment-max into mord[dst]
__global__ __launch_bounds__(256) void edge_score_max_kernel(
    const int* __restrict__ ei, int E, int Ep,
    const float* __restrict__ ssrc, const float* __restrict__ sdst,
    float* __restrict__ ebuf, unsigned* __restrict__ mord, int H) {
  int idx = blockIdx.x * blockDim.x + threadIdx.x;
  if (idx >= Ep * H) return;
  int e = idx / H, hh = idx - e * H;
  int s, d; edge_endpoints(ei, E, e, s, d);
  float v = ssrc[s * H + hh] + sdst[d * H + hh];
  v = (v > 0.f) ? v : NEG_SLOPE * v;
  ebuf[idx] = v;
  atomicMax(&mord[d * H + hh], ord_f32(v));
}

// Pass 2: ex = exp(e - m[dst]); store; segment-sum into denom[dst]
__global__ __launch_bounds__(256) void edge_exp_sum_kernel(
    const int* __restrict__ ei, int E, int Ep,
    float* __restrict__ ebuf, const unsigned* __restrict__ mord,
    float* __restrict__ denom, int H) {
  int idx = blockIdx.x * blockDim.x + threadIdx.x;
  if (idx >= Ep * H) return;
  int e = idx / H, hh = idx - e * H;
  int s, d; edge_endpoints(ei, E, e, s, d);
  (void)s;
  float ex = expf(ebuf[idx] - unord_f32(mord[d * H + hh]));
  ebuf[idx] = ex;
  atomicAdd(&denom[d * H + hh], ex);
}

// Pass 3: out[dst,h,:] += h[src,h,:] * (ex / (denom[dst,h]+1e-16)), float4 per thread
__global__ __launch_bounds__(256) void edge_aggregate_kernel(
    const int* __restrict__ ei, int E, int Ep,
    const float* __restrict__ ebuf, const float* __restrict__ denom,
    const float* __restrict__ h, float* __restrict__ out, int H, int C) {
  long long idx = (long long)blockIdx.x * blockDim.x + threadIdx.x;
  const int cq = C >> 2;
  const int q  = H * cq;
  long long total = (long long)Ep * q;
  if (idx >= total) return;
  int e   = (int)(idx / q);
  int rem = (int)(idx - (long long)e * q);
  int hh  = rem / cq;
  int c4  = (rem - hh * cq) * 4;
  int s, d; edge_endpoints(ei, E, e, s, d);
  float alpha = ebuf[(size_t)e * H + hh] / (denom[d * H + hh] + 1e-16f);
  const float4 hv = *(const float4*)(h + ((size_t)s * H + hh) * C + c4);
  float* op = out + ((size_t)d * H + hh) * C + c4;
  atomicAdd(op + 0, hv.x * alpha);
  atomicAdd(op + 1, hv.y * alpha);
  atomicAdd(op + 2, hv.z * alpha);
  atomicAdd(op + 3, hv.w * alpha);
}

__global__ __launch_bounds__(256) void bias_relu_kernel(
    float* __restrict__ x, const float* __restrict__ b, long long total, int F) {
  long long idx = (long long)blockIdx.x * blockDim.x + threadIdx.x;
  if (idx >= total) return;
  int f = (int)(idx % F);
  float v = x[idx] + b[f];
  x[idx] = (v > 0.f) ? v : 0.f;
}

__global__ __launch_bounds__(256) void pool_kernel(
    const float* __restrict__ h, const int* __restrict__ batch,
    float* __restrict__ pooled, float* __restrict__ counts, int N, int C) {
  int idx = blockIdx.x * blockDim.x + threadIdx.x;
  if (idx >= N * C) return;
  int n = idx / C, f = idx - n * C;
  int g = batch[n];
  atomicAdd(&pooled[(size_t)g * C + f], h[idx]);
  if (f == 0) atomicAdd(&counts[g], 1.0f);
}

__global__ __launch_bounds__(256) void classifier_kernel(
    const float* __restrict__ pooled, const float* __restrict__ counts,
    const float* __restrict__ Wc, const float* __restrict__ bc,
    float* __restrict__ out, int G, int C, int CLS) {
  int idx = blockIdx.x * blockDim.x + threadIdx.x;
  if (idx >= G * CLS) return;
  int g = idx / CLS, c = idx - g * CLS;
  float inv = 1.0f / fmaxf(counts[g], 1.0f);
  float s = 0.f;
  for (int k = 0; k < C; ++k) s += pooled[(size_t)g * C + k] * Wc[k * CLS + c];
  out[idx] = s * inv + bc[c];
}

// ---------------------------------------------------------------------------
extern "C" void kernel_launch(void* const* d_in, const int* in_sizes, int n_in,
                              void* d_out, int out_size, void* d_ws, size_t ws_size,
                              hipStream_t stream) {
  const float* x      = (const float*)d_in[0];
  const int*   ei     = (const int*)d_in[1];
  const int*   batch  = (const int*)d_in[2];
  const float* W1     = (const float*)d_in[3];
  const float* a_src1 = (const float*)d_in[4];
  const float* a_dst1 = (const float*)d_in[5];
  const float* b1     = (const float*)d_in[6];
  const float* W2     = (const float*)d_in[7];
  const float* a_src2 = (const float*)d_in[8];
  const float* a_dst2 = (const float*)d_in[9];
  const float* b2     = (const float*)d_in[10];
  const float* Wc     = (const float*)d_in[11];
  const float* bc     = (const float*)d_in[12];
  float* out = (float*)d_out;

  const int F_IN = 128, HID = 64, HEADS = 4, G = 256, CLS = 10;
  const int F1 = HEADS * HID;                  // 256
  const int N  = in_sizes[0] / F_IN;           // 50000
  const int E  = in_sizes[1] / 2;              // 800000
  const int Ep = E + N;                        // edges + self loops

  // ---- workspace layout (fp32 elements); layer-2 reuses the h1 region ----
  float* ws = (float*)d_ws;
  float*    h1     = ws;                               // N*F1
  float*    out1   = h1   + (size_t)N * F1;            // N*F1
  float*    ssrc   = out1 + (size_t)N * F1;            // N*HEADS
  float*    sdst   = ssrc + (size_t)N * HEADS;         // N*HEADS
  float*    denom  = sdst + (size_t)N * HEADS;         // N*HEADS
  unsigned* mord   = (unsigned*)(denom + (size_t)N * HEADS);  // N*HEADS
  float*    ebuf   = (float*)(mord + (size_t)N * HEADS);      // Ep*HEADS
  float*    pooled = ebuf + (size_t)Ep * HEADS;        // G*HID
  float*    counts = pooled + (size_t)G * HID;         // G
  float*    h2     = h1;                               // reuse (h1 dead after L1 agg)
  float*    out2   = h1 + (size_t)N * HID;

  const int TB = 256;
  const int mtiles  = N / 16;                          // 50000 % 16 == 0
  const dim3 gemmB(128);

  // ======================= Layer 1 (4 heads, C=64) =======================
  gemm_wmma_f32<<<dim3((mtiles + 3) / 4, F1 / 64), gemmB, 0, stream>>>(
      x, W1, h1, N, F_IN, F1);
  scores_kernel<<<(N * HEADS + TB - 1) / TB, TB, 0, stream>>>(
      h1, a_src1, a_dst1, ssrc, sdst, N, HEADS, HID);
  hipMemsetAsync(mord,  0, (size_t)N * HEADS * sizeof(unsigned), stream);
  hipMemsetAsync(denom, 0, (size_t)N * HEADS * sizeof(float), stream);
  hipMemsetAsync(out1,  0, (size_t)N * F1 * sizeof(float), stream);
  edge_score_max_kernel<<<(Ep * HEADS + TB - 1) / TB, TB, 0, stream>>>(
      ei, E, Ep, ssrc, sdst, ebuf, mord, HEADS);
  edge_exp_sum_kernel<<<(Ep * HEADS + TB - 1) / TB, TB, 0, stream>>>(
      ei, E, Ep, ebuf, mord, denom, HEADS);
  {
    long long tot = (long long)Ep * HEADS * (HID / 4);
    edge_aggregate_kernel<<<(unsigned)((tot + TB - 1) / TB), TB, 0, stream>>>(
        ei, E, Ep, ebuf, denom, h1, out1, HEADS, HID);
  }
  bias_relu_kernel<<<(unsigned)(((long long)N * F1 + TB - 1) / TB), TB, 0, stream>>>(
      out1, b1, (long long)N * F1, F1);

  // ======================= Layer 2 (1 head, C=64) ========================
  gemm_wmma_f32<<<dim3((mtiles + 3) / 4, HID / 64), gemmB, 0, stream>>>(
      out1, W2, h2, N, F1, HID);
  hipMemsetAsync(mord,  0, (size_t)N * sizeof(unsigned), stream);
  hipMemsetAsync(denom, 0, (size_t)N * sizeof(float), stream);
  hipMemsetAsync(out2,  0, (size_t)N * HID * sizeof(float), stream);
  scores_kernel<<<(N + TB - 1) / TB, TB, 0, stream>>>(
      h2, a_src2, a_dst2, ssrc, sdst, N, 1, HID);
  edge_score_max_kernel<<<(Ep + TB - 1) / TB, TB, 0, stream>>>(
      ei, E, Ep, ssrc, sdst, ebuf, mord, 1);
  edge_exp_sum_kernel<<<(Ep + TB - 1) / TB, TB, 0, stream>>>(
      ei, E, Ep, ebuf, mord, denom, 1);
  {
    long long tot = (long long)Ep * (HID / 4);
    edge_aggregate_kernel<<<(unsigned)((tot + TB - 1) / TB), TB, 0, stream>>>(
        ei, E, Ep, ebuf, denom, h2, out2, 1, HID);
  }
  bias_relu_kernel<<<(unsigned)(((long long)N * HID + TB - 1) / TB), TB, 0, stream>>>(
      out2, b2, (long long)N * HID, HID);

  // ==================== Global mean pool + classifier ====================
  hipMemsetAsync(pooled, 0, (size_t)(G * HID + G) * sizeof(float), stream);
  pool_kernel<<<(N * HID + TB - 1) / TB, TB, 0, stream>>>(
      out2, batch, pooled, counts, N, HID);
  classifier_kernel<<<(G * CLS + TB - 1) / TB, TB, 0, stream>>>(
      pooled, counts, Wc, bc, out, G, HID, CLS);
}